// MolecularGCN_49993419325830
// MI455X (gfx1250) — compile-verified
//
#include <hip/hip_runtime.h>

typedef __attribute__((ext_vector_type(2))) float v2f;
typedef __attribute__((ext_vector_type(8))) float v8f;

#define DIM 64
#define BN_EPS 1e-5f

// ---------------------------------------------------------------- utilities
__global__ void zero_kernel(float* __restrict__ p, long long n) {
  long long i = (long long)blockIdx.x * blockDim.x + threadIdx.x;
  if (i < n) p[i] = 0.0f;
}

__global__ void deg_accum_kernel(const int* __restrict__ dst, float* __restrict__ deg, int E) {
  int e = blockIdx.x * blockDim.x + threadIdx.x;
  if (e < E) atomicAdd(&deg[dst[e]], 1.0f);
}

__global__ void dis_kernel(float* __restrict__ deg_dis, int N) {
  int i = blockIdx.x * blockDim.x + threadIdx.x;
  if (i < N) deg_dis[i] = rsqrtf(deg_dis[i] + 1.0f);  // D^{-1/2} with self-loop
}

__global__ void enorm_kernel(const int* __restrict__ src, const int* __restrict__ dst,
                             const float* __restrict__ dis, float* __restrict__ enorm, int E) {
  int e = blockIdx.x * blockDim.x + threadIdx.x;
  if (e < E) enorm[e] = dis[src[e]] * dis[dst[e]];
}

// ------------------------------------------------- f32 WMMA GEMM: C = A @ W (+bias, relu)
// A: [M x K] row-major, W: [K x 64] row-major, C: [M x 64].
// M must be a multiple of 32 (100000 and 4096 both are). K multiple of 4.
// 8 waves / 256-thread block; wave w owns the 16x16 tile (rowTile=w>>2, colTile=w&3).
// Fragment layout per CDNA5 ISA (32-bit A 16x4): lanes 0-15 carry K=0,1; lanes 16-31 carry K=2,3.
template <int K>
__global__ __launch_bounds__(256) void gemm_wmma_f32(
    const float* __restrict__ A, const float* __restrict__ W,
    float* __restrict__ C, const float* __restrict__ bias, int relu) {
  const int wave  = threadIdx.x >> 5;
  const int lane  = threadIdx.x & 31;
  const int lhalf = lane >> 4;   // 0: K=0,1 ; 1: K=2,3
  const int lmod  = lane & 15;
  const int row0  = blockIdx.x * 32 + (wave >> 2) * 16;
  const int col0  = (wave & 3) * 16;

  v8f acc = {0.f, 0.f, 0.f, 0.f, 0.f, 0.f, 0.f, 0.f};
  const float* ap = A + (size_t)(row0 + lmod) * K + lhalf * 2;   // row = row0+lmod
  const float* bp = W + (size_t)(lhalf * 2) * DIM + col0 + lmod; // col = col0+lmod

#pragma unroll
  for (int k = 0; k < K; k += 4) {
    v2f a; a.x = ap[k];               a.y = ap[k + 1];               // A[row][k+2*lhalf +0/1]
    v2f b; b.x = bp[(size_t)k * DIM]; b.y = bp[(size_t)(k + 1) * DIM]; // W[k+2*lhalf +0/1][col]
    acc = __builtin_amdgcn_wmma_f32_16x16x4_f32(false, a, false, b, (short)0, acc, false, false);
  }

  // C/D layout: VGPR v, lanes 0-15 -> M=v, lanes 16-31 -> M=v+8 (N = lane%16)
  const int col = col0 + lmod;
  const float bv = bias ? bias[col] : 0.0f;
  const int rbase = row0 + lhalf * 8;
#pragma unroll
  for (int v = 0; v < 8; ++v) {
    float val = acc[v] + bv;
    if (relu) val = fmaxf(val, 0.0f);
    C[(size_t)(rbase + v) * DIM + col] = val;
  }
}

// -------------------------------------------------------------- edge scatter
// agg[dst] += h[src] * enorm[e]; 16 threads per edge, float4 per thread.
__global__ void scatter_kernel(const float* __restrict__ h, const int* __restrict__ src,
                               const int* __restrict__ dst, const float* __restrict__ enorm,
                               float* __restrict__ agg, int E) {
  long long t = (long long)blockIdx.x * blockDim.x + threadIdx.x;
  if (t >= (long long)E * 16) return;
  int e = (int)(t >> 4);
  int q = ((int)t & 15) * 4;
  int s = src[e], d = dst[e];
  float w = enorm[e];
  float4 v = *(const float4*)(h + (size_t)s * DIM + q);
  float* p = agg + (size_t)d * DIM + q;
  atomicAdd(p + 0, v.x * w);
  atomicAdd(p + 1, v.y * w);
  atomicAdd(p + 2, v.z * w);
  atomicAdd(p + 3, v.w * w);
}

// out = agg + h * dis[i]^2 + bias  (+ optional relu)
__global__ void combine_kernel(const float* __restrict__ h, const float* __restrict__ agg,
                               const float* __restrict__ dis, const float* __restrict__ bias,
                               float* __restrict__ outp, int N, int relu) {
  long long t = (long long)blockIdx.x * blockDim.x + threadIdx.x;
  if (t >= (long long)N * 16) return;
  int node = (int)(t >> 4);
  int c = ((int)t & 15) * 4;
  float sn = dis[node]; sn *= sn;  // self-loop norm = 1/deg
  float4 hv = *(const float4*)(h + (size_t)node * DIM + c);
  float4 av = *(const float4*)(agg + (size_t)node * DIM + c);
  float4 r;
  r.x = av.x + hv.x * sn + bias[c + 0];
  r.y = av.y + hv.y * sn + bias[c + 1];
  r.z = av.z + hv.z * sn + bias[c + 2];
  r.w = av.w + hv.w * sn + bias[c + 3];
  if (relu) {
    r.x = fmaxf(r.x, 0.f); r.y = fmaxf(r.y, 0.f);
    r.z = fmaxf(r.z, 0.f); r.w = fmaxf(r.w, 0.f);
  }
  *(float4*)(outp + (size_t)node * DIM + c) = r;
}

// ---------------------------------------------------------------- batchnorm
__global__ void bn_reduce_kernel(const float* __restrict__ h, int n, float* __restrict__ sums) {
  __shared__ float ssum[256];
  __shared__ float ssq[256];
  int col = threadIdx.x & 63;
  int sub = threadIdx.x >> 6;  // 0..3
  float a = 0.f, a2 = 0.f;
  for (int r = blockIdx.x * 4 + sub; r < n; r += gridDim.x * 4) {
    float v = h[(size_t)r * DIM + col];
    a += v; a2 += v * v;
  }
  ssum[threadIdx.x] = a; ssq[threadIdx.x] = a2;
  __syncthreads();
  if (sub == 0) {
    a  = ssum[col] + ssum[col + 64] + ssum[col + 128] + ssum[col + 192];
    a2 = ssq[col]  + ssq[col + 64]  + ssq[col + 128]  + ssq[col + 192];
    atomicAdd(&sums[col], a);
    atomicAdd(&sums[64 + col], a2);
  }
}

__global__ void bn_finalize_kernel(const float* __restrict__ sums, const float* __restrict__ g,
                                   const float* __restrict__ b, float inv_n,
                                   float* __restrict__ ss) {
  int c = threadIdx.x;
  if (c >= 64) return;
  float mean = sums[c] * inv_n;
  float var  = sums[64 + c] * inv_n - mean * mean;  // biased variance (jnp.var)
  float scale = g[c] * rsqrtf(var + BN_EPS);
  ss[c] = scale;
  ss[64 + c] = b[c] - mean * scale;
}

__global__ void bn_apply_kernel(float* __restrict__ h, const float* __restrict__ ss, long long nq) {
  long long t = (long long)blockIdx.x * blockDim.x + threadIdx.x;
  if (t >= nq) return;
  int c = ((int)t & 15) * 4;
  float4 v = ((float4*)h)[t];
  v.x = v.x * ss[c + 0] + ss[64 + c + 0];
  v.y = v.y * ss[c + 1] + ss[64 + c + 1];
  v.z = v.z * ss[c + 2] + ss[64 + c + 2];
  v.w = v.w * ss[c + 3] + ss[64 + c + 3];
  ((float4*)h)[t] = v;
}

// -------------------------------------------------------------------- pool
__global__ void pool_kernel(const float* __restrict__ h, const int* __restrict__ batch,
                            float* __restrict__ p, int N) {
  long long t = (long long)blockIdx.x * blockDim.x + threadIdx.x;
  if (t >= (long long)N * 16) return;
  int node = (int)(t >> 4);
  int q = ((int)t & 15) * 4;
  int g = batch[node];
  float4 v = *(const float4*)(h + (size_t)node * DIM + q);
  float* pp = p + (size_t)g * DIM + q;
  atomicAdd(pp + 0, v.x);
  atomicAdd(pp + 1, v.y);
  atomicAdd(pp + 2, v.z);
  atomicAdd(pp + 3, v.w);
}

// ---------------------------------------------------------- final projection
__global__ void out_kernel(const float* __restrict__ p, const float* __restrict__ w,
                           const float* __restrict__ b, float* __restrict__ out, int G) {
  int g = blockIdx.x * blockDim.x + threadIdx.x;
  if (g >= G) return;
  const float4* pv = (const float4*)(p + (size_t)g * DIM);
  const float4* wv = (const float4*)w;
  float acc = 0.f;
#pragma unroll
  for (int i = 0; i < 16; ++i) {
    float4 a = pv[i], c = wv[i];
    acc += a.x * c.x + a.y * c.y + a.z * c.z + a.w * c.w;
  }
  out[g] = acc + b[0];
}

// ==================================================================== launch
extern "C" void kernel_launch(void* const* d_in, const int* in_sizes, int n_in,
                              void* d_out, int out_size, void* d_ws, size_t ws_size,
                              hipStream_t stream) {
  const float* x      = (const float*)d_in[0];
  const int*   edge   = (const int*)d_in[1];   // [2,E]: src = edge[0:E), dst = edge[E:2E)
  const int*   batch  = (const int*)d_in[2];
  const float* w1     = (const float*)d_in[3];
  const float* b1     = (const float*)d_in[4];
  const float* bn1_g  = (const float*)d_in[5];
  const float* bn1_b  = (const float*)d_in[6];
  const float* conv_w = (const float*)d_in[7];
  const float* conv_b = (const float*)d_in[8];
  const float* bnc_g  = (const float*)d_in[9];
  const float* bnc_b  = (const float*)d_in[10];
  const float* mlp_w  = (const float*)d_in[11];
  const float* mlp_b  = (const float*)d_in[12];
  const float* bnm_g  = (const float*)d_in[13];
  const float* bnm_b  = (const float*)d_in[14];
  const float* out_w  = (const float*)d_in[15];
  const float* out_b  = (const float*)d_in[16];
  float* out = (float*)d_out;
  (void)n_in; (void)ws_size;

  const int N = in_sizes[0] / 128;   // 100000 (multiple of 32)
  const int E = in_sizes[1] / 2;     // 3200000
  const int G = out_size;            // 4096   (multiple of 32)
  const int* src = edge;
  const int* dst = edge + E;

  // workspace layout
  float* ws    = (float*)d_ws;
  float* dis   = ws;                          // N
  float* hA    = dis + N;                     // N*64  (GEMM output)
  float* agg   = hA + (size_t)N * DIM;        // N*64  (edge aggregation)
  float* hB    = agg + (size_t)N * DIM;       // N*64  (layer activations)
  float* enorm = hB + (size_t)N * DIM;        // E
  float* pA    = enorm + E;                   // G*64
  float* pB    = pA + (size_t)G * DIM;        // G*64
  float* sums  = pB + (size_t)G * DIM;        // 128 (sum, sumsq)
  float* ssbuf = sums + 128;                  // 128 (scale, shift)

  const int T = 256;
  auto blocks = [&](long long n) { return (int)((n + T - 1) / T); };

  auto run_bn = [&](float* h, int n, const float* g, const float* b) {
    zero_kernel<<<1, 128, 0, stream>>>(sums, 128);
    bn_reduce_kernel<<<512, T, 0, stream>>>(h, n, sums);
    bn_finalize_kernel<<<1, 64, 0, stream>>>(sums, g, b, 1.0f / (float)n, ssbuf);
    long long nq = (long long)n * 16;
    bn_apply_kernel<<<blocks(nq), T, 0, stream>>>(h, ssbuf, nq);
  };

  auto run_gcn = [&](const float* in, int K, const float* W, const float* bias, int relu) {
    if (K == 128) gemm_wmma_f32<128><<<N / 32, T, 0, stream>>>(in, W, hA, nullptr, 0);
    else          gemm_wmma_f32<64><<<N / 32, T, 0, stream>>>(in, W, hA, nullptr, 0);
    zero_kernel<<<blocks((long long)N * DIM), T, 0, stream>>>(agg, (long long)N * DIM);
    scatter_kernel<<<blocks((long long)E * 16), T, 0, stream>>>(hA, src, dst, enorm, agg, E);
    combine_kernel<<<blocks((long long)N * 16), T, 0, stream>>>(hA, agg, dis, bias, hB, N, relu);
  };

  // ---- degree / norms ----
  zero_kernel<<<blocks(N), T, 0, stream>>>(dis, N);
  deg_accum_kernel<<<blocks(E), T, 0, stream>>>(dst, dis, E);
  dis_kernel<<<blocks(N), T, 0, stream>>>(dis, N);
  enorm_kernel<<<blocks(E), T, 0, stream>>>(src, dst, dis, enorm, E);

  // ---- GCN stack ----
  run_gcn(x, 128, w1, b1, /*relu=*/1);
  run_bn(hB, N, bn1_g, bn1_b);
  for (int i = 0; i < 3; ++i) {
    run_gcn(hB, 64, conv_w + (size_t)i * DIM * DIM, conv_b + i * DIM, /*relu=*/0);
    run_bn(hB, N, bnc_g + i * DIM, bnc_b + i * DIM);
  }

  // ---- pool ----
  zero_kernel<<<blocks((long long)G * DIM), T, 0, stream>>>(pA, (long long)G * DIM);
  pool_kernel<<<blocks((long long)N * 16), T, 0, stream>>>(hB, batch, pA, N);

  // ---- MLP head ----
  float* cur = pA;
  float* nxt = pB;
  for (int i = 0; i < 3; ++i) {
    gemm_wmma_f32<64><<<G / 32, T, 0, stream>>>(cur, mlp_w + (size_t)i * DIM * DIM, nxt,
                                                mlp_b + i * DIM, /*relu=*/1);
    run_bn(nxt, G, bnm_g + i * DIM, bnm_b + i * DIM);
    float* t2 = cur; cur = nxt; nxt = t2;
  }

  // ---- output projection ----
  out_kernel<<<(G + 63) / 64, 64, 0, stream>>>(cur, out_w, out_b, out, G);
}